// SmallerGCNConvNet_16561393893733
// MI455X (gfx1250) — compile-verified
//
#include <hip/hip_runtime.h>
#include <math.h>

#define NNODES 100000     // multiple of 16 -> no row guards in GEMM
#define NEDGES 3200000
#define BN_EPS 1e-5f
#define MAXPAD 52         // cpad(50)

typedef float v2f __attribute__((ext_vector_type(2)));
typedef float v8f __attribute__((ext_vector_type(8)));

// ---------------- degree / dinv ----------------
__global__ void k_deg_init(float* __restrict__ deg, int n) {
  int i = blockIdx.x * blockDim.x + threadIdx.x;
  if (i < n) deg[i] = 1.0f;  // +1 self loop
}

__global__ void k_deg_count(const int* __restrict__ ei, float* __restrict__ deg, int ne) {
  int e = blockIdx.x * blockDim.x + threadIdx.x;
  if (e < ne) atomicAdd(&deg[ei[ne + e]], 1.0f);  // dst = row 1 of edge_index
}

__global__ void k_dinv(float* __restrict__ deg, int n) {
  int i = blockIdx.x * blockDim.x + threadIdx.x;
  if (i < n) deg[i] = rsqrtf(deg[i]);
}

// ---------------- pad x [N,50] -> bufH [N,52] (zero pad cols) ----------------
__global__ void k_pad_input(const float* __restrict__ X, float* __restrict__ H, int n) {
  unsigned int idx = blockIdx.x * blockDim.x + threadIdx.x;
  unsigned int total = (unsigned int)n * MAXPAD;
  if (idx >= total) return;
  int i = idx / MAXPAD;
  int c = idx - i * MAXPAD;
  H[idx] = (c < 50) ? X[i * 50 + c] : 0.f;
}

// ---------------- WMMA GEMM: HW[n,sout] = H[n,CINP] @ W[cin,cout] ----------------
// CINP = padded K (multiple of 4, compile-time -> fully unrolled WMMA chain).
// W staged transposed+zero-padded into LDS: Wlds[col*CINP + k].
// One wave32 per 16x16 output tile. N multiple of 16 -> no row guards.
template <int CINP>
__global__ void k_gemm_wmma(const float* __restrict__ H, const float* __restrict__ Wm,
                            float* __restrict__ HW, int cin, int cout, int sout,
                            int cols, int col_tiles, int total_waves) {
  __shared__ float Wlds[64 * CINP];   // cols <= 64

  // cooperative staging: Wlds[c][k] = W[k][c], zero-padded
  for (int idx = threadIdx.x; idx < cols * CINP; idx += blockDim.x) {
    int c = idx / CINP;
    int k = idx - c * CINP;
    Wlds[idx] = (c < cout && k < cin) ? Wm[k * cout + c] : 0.f;
  }
  __syncthreads();

  int wave = (blockIdx.x * blockDim.x + threadIdx.x) >> 5;
  if (wave >= total_waves) return;             // wave-uniform
  int rt = wave / col_tiles;
  int ct = wave - rt * col_tiles;
  int lane = threadIdx.x & 31;
  int half = lane >> 4;                        // 0: K+0..1, 1: K+2..3
  int m    = lane & 15;

  const float* Arow = H + (size_t)(rt * 16 + m) * CINP + half * 2;
  const float* Bcol = Wlds + (ct * 16 + m) * CINP + half * 2;

  v8f acc = {0.f, 0.f, 0.f, 0.f, 0.f, 0.f, 0.f, 0.f};
#pragma unroll
  for (int k0 = 0; k0 < CINP; k0 += 4) {
    v2f a = *(const v2f*)(Arow + k0);          // global_load_b64, no guards
    v2f b = *(const v2f*)(Bcol + k0);          // ds_load_b64, no guards
    acc = __builtin_amdgcn_wmma_f32_16x16x4_f32(false, a, false, b, (short)0, acc,
                                                false, false);
  }

  int bcol = ct * 16 + m;
  if (bcol < sout) {                           // pad cols [cout,sout) get zeros (B cols are 0)
    float* out = HW + (size_t)(rt * 16 + half * 8) * sout + bcol;
#pragma unroll
    for (int r = 0; r < 8; ++r) out[(size_t)r * sout] = acc[r];
  }
}

// ---------------- edge scatter: agg[dst] += hw[src] * dinv[src] ----------------
__global__ void k_edge_scatter(const int* __restrict__ ei, const float* __restrict__ HW,
                               const float* __restrict__ dinv, float* __restrict__ AGG,
                               int ne, int cout, int s) {
  unsigned int idx = blockIdx.x * blockDim.x + threadIdx.x;
  unsigned int total = (unsigned int)ne * (unsigned int)cout;
  if (idx >= total) return;
  int e = idx / cout;
  int c = idx - e * cout;
  int sn = ei[e];        // src
  int dn = ei[ne + e];   // dst
  float v = HW[sn * s + c] * dinv[sn];
  atomicAdd(&AGG[dn * s + c], v);
}

// ---------------- finalize: out = agg*dinv + hw*dinv^2 + b (opt. ELU) ----------------
// HW/AGG have stride sin; OUT has stride sout (pad cols of OUT get 0).
__global__ void k_finalize(const float* __restrict__ HW, const float* __restrict__ AGG,
                           const float* __restrict__ dinv, const float* __restrict__ bias,
                           float* __restrict__ OUT, int n, int cout, int sin, int sout,
                           int apply_elu) {
  unsigned int idx = blockIdx.x * blockDim.x + threadIdx.x;
  unsigned int total = (unsigned int)n * (unsigned int)sout;
  if (idx >= total) return;
  int i = idx / sout;
  int c = idx - i * sout;
  if (c < cout) {
    float di = dinv[i];
    unsigned int a = (unsigned int)i * sin + c;
    float v = AGG[a] * di + HW[a] * di * di + bias[c];
    if (apply_elu) v = (v > 0.f) ? v : expm1f(v);
    OUT[idx] = v;
  } else {
    OUT[idx] = 0.f;      // keep pad columns zero for next layer's A loads
  }
}

// ---------------- BatchNorm stats (one block per channel) ----------------
__global__ void k_bn_stats(const float* __restrict__ H, float* __restrict__ mv,
                           int n, int s) {
  int c = blockIdx.x;
  float sum = 0.f, sum2 = 0.f;
  for (int i = threadIdx.x; i < n; i += blockDim.x) {
    float v = H[(size_t)i * s + c];
    sum += v; sum2 += v * v;
  }
  __shared__ float sh[256], sh2[256];
  sh[threadIdx.x] = sum; sh2[threadIdx.x] = sum2;
  __syncthreads();
  for (int o = 128; o > 0; o >>= 1) {
    if ((int)threadIdx.x < o) {
      sh[threadIdx.x]  += sh[threadIdx.x + o];
      sh2[threadIdx.x] += sh2[threadIdx.x + o];
    }
    __syncthreads();
  }
  if (threadIdx.x == 0) {
    float m = sh[0] / (float)n;
    mv[c]          = m;
    mv[MAXPAD + c] = sh2[0] / (float)n - m * m;  // biased var (matches jnp.var)
  }
}

__global__ void k_bn_apply(float* __restrict__ H, const float* __restrict__ mv,
                           const float* __restrict__ g, const float* __restrict__ be,
                           int n, int cout, int s) {
  unsigned int idx = blockIdx.x * blockDim.x + threadIdx.x;
  unsigned int total = (unsigned int)n * (unsigned int)s;
  if (idx >= total) return;
  int i = idx / s;
  int c = idx - i * s;
  if (c >= cout) return;  // pad columns stay zero
  float m = mv[c];
  float v = mv[MAXPAD + c];
  H[idx] = (H[idx] - m) * rsqrtf(v + BN_EPS) * g[c] + be[c];
}

// ---------------- host orchestration ----------------
static inline size_t align256(size_t x) { return (x + 255) & ~(size_t)255; }
static inline int cpad(int x) { return (x + 3) & ~3; }

extern "C" void kernel_launch(void* const* d_in, const int* in_sizes, int n_in,
                              void* d_out, int out_size, void* d_ws, size_t ws_size,
                              hipStream_t stream) {
  (void)in_sizes; (void)n_in; (void)out_size; (void)ws_size;
  const int DIMS[13] = {50, 40, 30, 20, 10, 10, 10, 10, 20, 30, 40, 50, 50};
  const int N = NNODES, E = NEDGES;
  const int BLK = 256;

  const float* x  = (const float*)d_in[0];
  const int*   ei = (const int*)d_in[1];   // [2, E]

  const float *Wp[12], *Bp[12], *Gp[12], *BEp[12];
  {
    int p = 2;
    for (int i = 0; i < 12; ++i) {
      Wp[i] = (const float*)d_in[p++];
      Bp[i] = (const float*)d_in[p++];
      if (i < 11) { Gp[i] = (const float*)d_in[p++]; BEp[i] = (const float*)d_in[p++]; }
      else        { Gp[i] = nullptr; BEp[i] = nullptr; }
    }
  }

  // workspace layout
  char* ws = (char*)d_ws;
  size_t off = 0;
  float* dinv = (float*)(ws + off);  off = align256(off + (size_t)N * 4);
  float* mv   = (float*)(ws + off);  off = align256(off + 2 * MAXPAD * 4);
  size_t bigN = align256((size_t)N * MAXPAD * 4);
  float* bufH   = (float*)(ws + off);  off += bigN;
  float* bufHW  = (float*)(ws + off);  off += bigN;
  float* bufAGG = (float*)(ws + off);  off += bigN;

  // degree -> dinv
  k_deg_init<<<(N + BLK - 1) / BLK, BLK, 0, stream>>>(dinv, N);
  k_deg_count<<<(E + BLK - 1) / BLK, BLK, 0, stream>>>(ei, dinv, E);
  k_dinv<<<(N + BLK - 1) / BLK, BLK, 0, stream>>>(dinv, N);

  // pad x into bufH (stride 52, zero pad cols)
  {
    unsigned int total = (unsigned int)N * MAXPAD;
    k_pad_input<<<(total + BLK - 1) / BLK, BLK, 0, stream>>>(x, bufH, N);
  }

  for (int L = 0; L < 12; ++L) {
    int cin  = DIMS[L],     cout = DIMS[L + 1];
    int sin  = cpad(cin);   int sout = cpad(cout);
    int col_tiles = (cout + 15) / 16;
    int cols = col_tiles * 16;
    int total_waves = (N / 16) * col_tiles;
    int grid = (total_waves * 32 + BLK - 1) / BLK;

    // hw = h @ W  (fully unrolled WMMA f32 chain)
    switch (sin) {
      case 52: k_gemm_wmma<52><<<grid, BLK, 0, stream>>>(bufH, Wp[L], bufHW, cin, cout, sout, cols, col_tiles, total_waves); break;
      case 40: k_gemm_wmma<40><<<grid, BLK, 0, stream>>>(bufH, Wp[L], bufHW, cin, cout, sout, cols, col_tiles, total_waves); break;
      case 32: k_gemm_wmma<32><<<grid, BLK, 0, stream>>>(bufH, Wp[L], bufHW, cin, cout, sout, cols, col_tiles, total_waves); break;
      case 20: k_gemm_wmma<20><<<grid, BLK, 0, stream>>>(bufH, Wp[L], bufHW, cin, cout, sout, cols, col_tiles, total_waves); break;
      default: k_gemm_wmma<12><<<grid, BLK, 0, stream>>>(bufH, Wp[L], bufHW, cin, cout, sout, cols, col_tiles, total_waves); break;
    }

    // agg = 0 (padded stride)
    hipMemsetAsync(bufAGG, 0, (size_t)N * sout * sizeof(float), stream);

    // scatter edges (exact E*cout work items, padded addressing)
    {
      unsigned int total = (unsigned int)E * (unsigned int)cout;
      k_edge_scatter<<<(total + BLK - 1) / BLK, BLK, 0, stream>>>(
          ei, bufHW, dinv, bufAGG, E, cout, sout);
    }

    // finalize (+ELU for layers 0..10); layer 11 -> d_out with exact stride
    {
      int ostride = (L == 11) ? cout : sout;
      float* outp = (L == 11) ? (float*)d_out : bufH;
      unsigned int total = (unsigned int)N * (unsigned int)ostride;
      k_finalize<<<(total + BLK - 1) / BLK, BLK, 0, stream>>>(
          bufHW, bufAGG, dinv, Bp[L], outp, N, cout, sout, ostride, (L < 11) ? 1 : 0);
    }

    if (L < 11) {
      k_bn_stats<<<cout, 256, 0, stream>>>(bufH, mv, N, sout);
      unsigned int total = (unsigned int)N * (unsigned int)sout;
      k_bn_apply<<<(total + BLK - 1) / BLK, BLK, 0, stream>>>(
          bufH, mv, Gp[L], BEp[L], N, cout, sout);
    }
  }
}